// SampledSoftmaxLayer_39951785787724
// MI455X (gfx1250) — compile-verified
//
#include <hip/hip_runtime.h>
#include <math.h>

#define NUM_CLASSES 1000000
#define DIM 64
#define BATCH 4096
#define NUM_SAMPLED 1024

typedef __attribute__((ext_vector_type(2))) float v2f;
typedef __attribute__((ext_vector_type(8))) float v8f;

// loss kernel: one block = 16 batch rows, 128 threads = 4 waves.
// wave w computes sampled_logits[16 rows][256 cols] via V_WMMA_F32_16X16X4_F32,
// then block-wide softmax reduction over the 1025 logits per row.
__global__ __launch_bounds__(128) void sampled_softmax_fused(
    const float* __restrict__ item,   // (DIM, NUM_CLASSES) row-major
    const float* __restrict__ user,   // (DIM, BATCH) row-major
    const int*   __restrict__ labels, // (BATCH,)
    const int*   __restrict__ sids,   // (NUM_SAMPLED,)
    float*       __restrict__ out)    // (BATCH,)
{
    extern __shared__ char smem_raw[];
    float* lds_logits = (float*)smem_raw;                 // 16 * 1024
    float* lds_true   = lds_logits + 16 * NUM_SAMPLED;    // 16
    int*   lds_label  = (int*)(lds_true + 16);            // 16
    float* lds_red    = (float*)(lds_label + 16);         // 128
    float* lds_rowmax = lds_red + 128;                    // 16

    const int tid  = threadIdx.x;       // 0..127
    const int wave = tid >> 5;          // 0..3
    const int lane = tid & 31;
    const int m0   = blockIdx.x * 16;   // batch-row base of this block

    if (tid < 16) lds_label[tid] = labels[m0 + tid];
    __syncthreads();

    const float inv_lognc = 1.0f / __logf((float)NUM_CLASSES + 1.0f);

    // ---- A panel: rows m0..m0+15, K=0..63, fp32, WMMA 16x4 A layout ----
    // lanes 0-15 : V0=K(4kk+0), V1=K(4kk+1) for M=lane
    // lanes 16-31: V0=K(4kk+2), V1=K(4kk+3) for M=lane-16
    const int mrow  = lane & 15;
    const int khalf = (lane >> 4) << 1;   // 0 or 2
    v2f a[16];
#pragma unroll
    for (int kk = 0; kk < 16; ++kk) {
        int k = 4 * kk + khalf;
        a[kk].x = user[(size_t)k       * BATCH + (m0 + mrow)];
        a[kk].y = user[(size_t)(k + 1) * BATCH + (m0 + mrow)];
    }

    const int mbase = (lane >> 4) << 3;   // C layout: VGPR r holds M=r (+8 for upper lanes)

    // ---- 16 N-tiles of 16 columns for this wave ----
    for (int t = 0; t < 16; ++t) {
        int n  = wave * 256 + t * 16 + mrow;    // this lane's column (B-N and C-N agree)
        int id = sids[n];

        v8f c = {};
#pragma unroll
        for (int kk = 0; kk < 16; ++kk) {
            int k = 4 * kk + khalf;
            v2f b;
            b.x = item[(size_t)k       * NUM_CLASSES + id];
            b.y = item[(size_t)(k + 1) * NUM_CLASSES + id];
            c = __builtin_amdgcn_wmma_f32_16x16x4_f32(
                    false, a[kk], false, b, (short)0, c, false, false);
        }

        // epilogue: samp_log_q correction + accidental-hit mask, spill tile to LDS
        float cid = (float)id;
        float slq = __logf((float)NUM_SAMPLED *
                           ((__logf(cid + 2.0f) - __logf(cid + 1.0f)) * inv_lognc));
#pragma unroll
        for (int r = 0; r < 8; ++r) {
            int m = mbase + r;
            float v = c[r] - slq;
            if (id == lds_label[m]) v -= 1.0e9f;
            lds_logits[m * NUM_SAMPLED + n] = v;
        }
    }

    // ---- true logits: 16 rows x 64-dim gathered dot (cheap, wave 0 lanes) ----
    if (tid < 16) {
        int lab = lds_label[tid];
        float s = 0.0f;
#pragma unroll 8
        for (int k = 0; k < DIM; ++k)
            s += user[(size_t)k * BATCH + (m0 + tid)] *
                 item[(size_t)k * NUM_CLASSES + lab];
        float cl  = (float)lab;
        float tlq = __logf((float)NUM_SAMPLED *
                           ((__logf(cl + 2.0f) - __logf(cl + 1.0f)) * inv_lognc));
        lds_true[tid] = s - tlq;
    }
    __syncthreads();

    // ---- row-wise max: 8 threads per row, 128 cols each ----
    {
        int r = tid & 15, chunk = tid >> 4;
        const float* row = &lds_logits[r * NUM_SAMPLED + chunk * 128];
        float mx = -3.0e38f;
        for (int j = 0; j < 128; ++j) mx = fmaxf(mx, row[j]);
        lds_red[r * 8 + chunk] = mx;
    }
    __syncthreads();
    if (tid < 16) {
        float mx = lds_true[tid];
        for (int c2 = 0; c2 < 8; ++c2) mx = fmaxf(mx, lds_red[tid * 8 + c2]);
        lds_rowmax[tid] = mx;
    }
    __syncthreads();

    // ---- row-wise sum of exp ----
    {
        int r = tid & 15, chunk = tid >> 4;
        const float* row = &lds_logits[r * NUM_SAMPLED + chunk * 128];
        float mx = lds_rowmax[r];
        float s = 0.0f;
        for (int j = 0; j < 128; ++j) s += __expf(row[j] - mx);
        lds_red[r * 8 + chunk] = s;
    }
    __syncthreads();

    // ---- loss = -(true - max - log(sumexp)) ----
    if (tid < 16) {
        float mx = lds_rowmax[tid];
        float s  = __expf(lds_true[tid] - mx);
        for (int c2 = 0; c2 < 8; ++c2) s += lds_red[tid * 8 + c2];
        out[m0 + tid] = -(lds_true[tid] - mx - __logf(s));
    }
}

extern "C" void kernel_launch(void* const* d_in, const int* in_sizes, int n_in,
                              void* d_out, int out_size, void* d_ws, size_t ws_size,
                              hipStream_t stream) {
    const float* item   = (const float*)d_in[0]; // (64, 1000000)
    const float* user   = (const float*)d_in[1]; // (64, 4096)
    const int*   labels = (const int*)d_in[2];   // (4096, 1) flat
    const int*   sids   = (const int*)d_in[3];   // (1024,)
    float*       out    = (float*)d_out;         // (4096, 1) flat

    size_t smem = (size_t)(16 * NUM_SAMPLED + 16 + 16 + 128 + 16) * sizeof(float);
    dim3 grid(BATCH / 16), block(128);
    hipLaunchKernelGGL(sampled_softmax_fused, grid, block, smem, stream,
                       item, user, labels, sids, out);
}